// MipRayMarcher2_16793322127410
// MI455X (gfx1250) — compile-verified
//
#include <hip/hip_runtime.h>
#include <math.h>

// ---------------------------------------------------------------------------
// MipRayMarcher2 for MI455X (gfx1250).
// Memory-bound streaming scan: uses CDNA5 async global->LDS DMA path
// (global_load_async_to_lds_b128 + s_wait_asynccnt) for data staging.
// ---------------------------------------------------------------------------

#define NSAMP 96
#define RAYS_PER_BLOCK 32
// LDS layout (words), padded per-ray strides (16B-aligned, reduce bank conflicts)
#define C_STRIDE_W 292   // 96*3 = 288 data words + 4 pad
#define S_STRIDE_W 100   // 96 data words + 4 pad
#define LDS_WORDS (RAYS_PER_BLOCK * (C_STRIDE_W + 2 * S_STRIDE_W))  // 15744 w = 62976 B

__device__ __forceinline__ unsigned lds_addr_of(const void* p) {
    // flat shared pointer = {shared aperture (hi 32), LDS byte offset (lo 32)}
    return (unsigned)(unsigned long long)p;
}

__device__ __forceinline__ void async_g2l_b128(unsigned lds_byte_addr, const void* gptr) {
    // VDST = LDS byte address VGPR, VADDR = 64-bit global address, tracked by ASYNCcnt
    asm volatile("global_load_async_to_lds_b128 %0, %1, off"
                 :: "v"(lds_byte_addr), "v"(gptr)
                 : "memory");
}

__device__ __forceinline__ void wait_async0() {
#if __has_builtin(__builtin_amdgcn_s_wait_asynccnt)
    __builtin_amdgcn_s_wait_asynccnt(0);
    asm volatile("" ::: "memory");
#else
    asm volatile("s_wait_asynccnt 0x0" ::: "memory");
#endif
}

// --------------------------- min/max of depths ------------------------------

__global__ void mm_init_kernel(unsigned* mm) {
    mm[0] = 0x7F800000u;  // +inf bits (running min, uint-ordered for positive floats)
    mm[1] = 0u;           // running max
}

__global__ void mm_reduce_kernel(const float* __restrict__ z, long n, unsigned* mm) {
    float lo = INFINITY, hi = -INFINITY;
    for (long i = (long)blockIdx.x * blockDim.x + threadIdx.x; i < n;
         i += (long)gridDim.x * blockDim.x) {
        float v = z[i];
        lo = fminf(lo, v);
        hi = fmaxf(hi, v);
    }
    for (int o = 16; o > 0; o >>= 1) {
        lo = fminf(lo, __shfl_xor(lo, o, 32));
        hi = fmaxf(hi, __shfl_xor(hi, o, 32));
    }
    if ((threadIdx.x & 31) == 0) {
        atomicMin(mm + 0, __float_as_uint(lo));  // depths > 0 -> bit order == float order
        atomicMax(mm + 1, __float_as_uint(hi));
    }
}

// ------------------------------- ray marcher --------------------------------

__global__ void __launch_bounds__(32)
marcher_kernel(const float* __restrict__ colors,    // [nRays,96,3]
               const float* __restrict__ densities, // [nRays,96]
               const float* __restrict__ depths,    // [nRays,96]
               const int* __restrict__ white_back,  // [1]
               const unsigned* __restrict__ mm,     // {min_bits, max_bits}
               float* __restrict__ out_rgb,         // [nRays,3]
               float* __restrict__ out_depth,       // [nRays]
               float* __restrict__ out_w)           // [nRays,95]
{
    __shared__ float smem[LDS_WORDS];
    float* sC = smem;
    float* sD = smem + RAYS_PER_BLOCK * C_STRIDE_W;
    float* sZ = sD + RAYS_PER_BLOCK * S_STRIDE_W;

    const int lane = threadIdx.x;  // 0..31, one ray per lane
    const long rayBase = (long)blockIdx.x * RAYS_PER_BLOCK;

    // ---- Stage the block's (contiguous) ray data into LDS via async DMA ----
    // colors: 32 rays * 96 * 3 floats = 2304 16B-chunks, 72 chunks/ray
    {
        const float* gC = colors + rayBase * (NSAMP * 3);
        unsigned base = lds_addr_of(sC);
        #pragma unroll 4
        for (int t = 0; t < 72; ++t) {
            unsigned c = (unsigned)(t * 32 + lane);   // coalesced: 512B per instr
            unsigned ray = c / 72u;
            unsigned off = c - ray * 72u;             // 16B chunk within ray
            async_g2l_b128(base + ray * (C_STRIDE_W * 4u) + off * 16u,
                           gC + (size_t)c * 4);
        }
    }
    // densities & depths: 32 rays * 96 floats = 768 chunks, 24 chunks/ray each
    {
        const float* gD = densities + rayBase * NSAMP;
        const float* gZ = depths + rayBase * NSAMP;
        unsigned baseD = lds_addr_of(sD);
        unsigned baseZ = lds_addr_of(sZ);
        #pragma unroll 4
        for (int t = 0; t < 24; ++t) {
            unsigned c = (unsigned)(t * 32 + lane);
            unsigned ray = c / 24u;
            unsigned off = c - ray * 24u;
            unsigned d = ray * (S_STRIDE_W * 4u) + off * 16u;
            async_g2l_b128(baseD + d, gD + (size_t)c * 4);
            async_g2l_b128(baseZ + d, gZ + (size_t)c * 4);
        }
    }
    wait_async0();  // single wave per block: no barrier needed beyond ASYNCcnt

    // ---- Sequential transmittance scan over this lane's ray ----
    const float* myC = sC + lane * C_STRIDE_W;
    const float* myD = sD + lane * S_STRIDE_W;
    const float* myZ = sZ + lane * S_STRIDE_W;

    const long rayG = rayBase + lane;
    float* wOut = out_w + rayG * (NSAMP - 1);

    float T = 1.0f;
    float accR = 0.f, accG = 0.f, accB = 0.f, accZ = 0.f, wsum = 0.f;
    float c0r = myC[0], c0g = myC[1], c0b = myC[2];
    float d0 = myD[0], z0 = myZ[0];

    #pragma unroll 5
    for (int s = 1; s < NSAMP; ++s) {
        float c1r = myC[s * 3 + 0];
        float c1g = myC[s * 3 + 1];
        float c1b = myC[s * 3 + 2];
        float d1 = myD[s];
        float z1 = myZ[s];

        float delta = z1 - z0;
        float dmid = 0.5f * (d0 + d1);
        float zmid = 0.5f * (z0 + z1);

        // softplus(dmid)
        float sp = (dmid > 20.f) ? dmid : log1pf(__expf(dmid));
        float alpha = 1.f - __expf(-sp * delta);

        float w = alpha * T;
        T *= (1.f - alpha + 1e-10f);

        accR += w * 0.5f * (c0r + c1r);
        accG += w * 0.5f * (c0g + c1g);
        accB += w * 0.5f * (c0b + c1b);
        accZ += w * zmid;
        wsum += w;
        wOut[s - 1] = w;

        c0r = c1r; c0g = c1g; c0b = c1b; d0 = d1; z0 = z1;
    }

    if (*white_back != 0) {
        float add = 1.f - wsum;
        accR += add; accG += add; accB += add;
    }

    // composite_depth: nan -> +inf, then clip to [min(depths), max(depths)]
    float dmin = __uint_as_float(mm[0]);
    float dmax = __uint_as_float(mm[1]);
    float cd = accZ;
    if (cd != cd) cd = INFINITY;
    cd = fminf(fmaxf(cd, dmin), dmax);

    out_rgb[rayG * 3 + 0] = accR;
    out_rgb[rayG * 3 + 1] = accG;
    out_rgb[rayG * 3 + 2] = accB;
    out_depth[rayG] = cd;
}

// ------------------------------- launcher -----------------------------------

extern "C" void kernel_launch(void* const* d_in, const int* in_sizes, int n_in,
                              void* d_out, int out_size, void* d_ws, size_t ws_size,
                              hipStream_t stream) {
    const float* colors = (const float*)d_in[0];
    const float* densities = (const float*)d_in[1];
    const float* depths = (const float*)d_in[2];
    const int* white_back = (const int*)d_in[3];

    long nz = (long)in_sizes[2];        // B*R*96 (depths elements)
    int nRays = (int)(nz / NSAMP);      // 65536

    float* out_rgb = (float*)d_out;
    float* out_depth = out_rgb + (size_t)nRays * 3;
    float* out_w = out_depth + (size_t)nRays;

    unsigned* mm = (unsigned*)d_ws;

    hipLaunchKernelGGL(mm_init_kernel, dim3(1), dim3(1), 0, stream, mm);
    hipLaunchKernelGGL(mm_reduce_kernel, dim3(1024), dim3(256), 0, stream,
                       depths, nz, mm);
    hipLaunchKernelGGL(marcher_kernel, dim3(nRays / RAYS_PER_BLOCK), dim3(32), 0, stream,
                       colors, densities, depths, white_back, mm,
                       out_rgb, out_depth, out_w);
}